// MMDLoss_66030827209267
// MI455X (gfx1250) — compile-verified
//
#include <hip/hip_runtime.h>

// Problem constants (reference: N=8192, D=2048, fp32)
#define N_DIM 8192
#define D_DIM 2048
#define GY    256                      // number of partial row-stripes
#define ROWS_PER_BLK (N_DIM / GY)      // 32 rows per block

typedef __attribute__((ext_vector_type(2))) float v2f;
typedef __attribute__((ext_vector_type(8))) float v8f;

// ---------------------------------------------------------------------------
// Kernel 1: bandwidth-bound streaming column-sum of (f_x - f_t).
// Grid: (D/1024, GY) blocks of 256 threads. Each thread owns one float4
// column group and accumulates a 32-row stripe in registers, then writes one
// partial row of the GY x D workspace. All loads are coalesced b128.
// ---------------------------------------------------------------------------
__global__ __launch_bounds__(256) void colsum_kernel(
    const float* __restrict__ fx, const float* __restrict__ ft,
    float* __restrict__ partials)
{
    const int cg = blockIdx.x * 256 + threadIdx.x;      // float4 column group
    const size_t row0  = (size_t)blockIdx.y * ROWS_PER_BLK;
    const size_t base  = row0 * (D_DIM / 4) + (size_t)cg;
    const float4* __restrict__ px = (const float4*)fx + base;
    const float4* __restrict__ pt = (const float4*)ft + base;

    float4 acc; acc.x = 0.f; acc.y = 0.f; acc.z = 0.f; acc.w = 0.f;
#pragma unroll 8
    for (int r = 0; r < ROWS_PER_BLK; ++r) {
        float4 x = px[(size_t)r * (D_DIM / 4)];
        float4 t = pt[(size_t)r * (D_DIM / 4)];
        acc.x += x.x - t.x;
        acc.y += x.y - t.y;
        acc.z += x.z - t.z;
        acc.w += x.w - t.w;
    }
    ((float4*)partials)[(size_t)blockIdx.y * (D_DIM / 4) + cg] = acc;
}

// ---------------------------------------------------------------------------
// Kernel 2: reduce GY x D partials -> ||s||^2 / N^2 using WMMA.
// For each 16-column tile, s[tile] = ones(16x4) @ P_chunk(4x16) accumulated
// over all GY rows via V_WMMA_F32_16X16X4_F32. With A == all-ones, every
// output row of D equals the column sums, so C-VGPR0 in lane l holds
// s[tile*16 + (l & 15)] (each column appearing twice across the 32 lanes;
// compensated by the final /2). One block, 16 waves, 8 tiles per wave.
// ---------------------------------------------------------------------------
__global__ __launch_bounds__(512) void finish_kernel(
    const float* __restrict__ partials, float* __restrict__ out)
{
    __shared__ float wsum[16];
    const int lane  = threadIdx.x & 31;
    const int wave  = threadIdx.x >> 5;          // 0..15
    const int nCol  = lane & 15;                 // N within tile
    const int kOff  = (lane < 16) ? 0 : 2;       // B-matrix K offset per half-wave

    v2f a; a.x = 1.0f; a.y = 1.0f;               // all-ones A (16x4)
    float sumsq = 0.0f;

    for (int t = 0; t < 8; ++t) {
        const int tile    = wave * 8 + t;        // 0..127
        const int colBase = tile * 16 + nCol;
        v8f c = {0.f, 0.f, 0.f, 0.f, 0.f, 0.f, 0.f, 0.f};
        for (int k = 0; k < GY; k += 4) {
            v2f b;
            b.x = partials[(size_t)(k + kOff)     * D_DIM + colBase];
            b.y = partials[(size_t)(k + kOff + 1) * D_DIM + colBase];
            // 8 args: (neg_a, A, neg_b, B, c_mod, C, reuse_a, reuse_b)
            c = __builtin_amdgcn_wmma_f32_16x16x4_f32(
                    false, a, false, b, (short)0, c, false, false);
        }
        const float s = c[0];                    // s[colBase], every lane
        sumsq += s * s;                          // each column counted twice
    }

    // wave32 reduction
    for (int off = 16; off >= 1; off >>= 1)
        sumsq += __shfl_xor(sumsq, off, 32);
    if (lane == 0) wsum[wave] = sumsq;
    __syncthreads();

    if (threadIdx.x == 0) {
        float tot = 0.0f;
#pragma unroll
        for (int w = 0; w < 16; ++w) tot += wsum[w];
        // /2 for the duplicated half-wave columns, /N^2 for the mean
        out[0] = (float)((double)tot / (2.0 * (double)N_DIM * (double)N_DIM));
    }
}

extern "C" void kernel_launch(void* const* d_in, const int* in_sizes, int n_in,
                              void* d_out, int out_size, void* d_ws, size_t ws_size,
                              hipStream_t stream) {
    const float* fx = (const float*)d_in[0];
    const float* ft = (const float*)d_in[1];
    float* out      = (float*)d_out;
    float* partials = (float*)d_ws;              // GY * D * 4 = 2 MB scratch

    dim3 g1(D_DIM / (256 * 4), GY);              // (2, 256) = 512 blocks
    colsum_kernel<<<g1, 256, 0, stream>>>(fx, ft, partials);
    finish_kernel<<<1, 512, 0, stream>>>(partials, out);
}